// Planes4D_7559142441480
// MI455X (gfx1250) — compile-verified
//
#include <hip/hip_runtime.h>

typedef float        v4f __attribute__((ext_vector_type(4)));
typedef unsigned int v4u __attribute__((ext_vector_type(4)));
typedef int          v4i __attribute__((ext_vector_type(4)));
typedef int          v8i __attribute__((ext_vector_type(8)));

static __device__ __forceinline__ int imin(int a, int b) { return a < b ? a : b; }

struct F8 { v4f lo, hi; };

// Bilinear interp of 8 channels from a (H, W, 8) channel-last plane.
// u01/v01 in [0,1); W = 1<<logW, H = 1<<logH.
static __device__ __forceinline__ F8 bilerp8(const float* __restrict__ base,
                                             int logW, int logH,
                                             float u01, float v01) {
  const int W = 1 << logW, H = 1 << logH;
  float u = fminf(fmaxf(u01 * (float)(W - 1), 0.0f), (float)(W - 1));
  float v = fminf(fmaxf(v01 * (float)(H - 1), 0.0f), (float)(H - 1));
  int x0 = (int)u;                 // u >= 0 -> trunc == floor
  int y0 = (int)v;
  int x1 = imin(x0 + 1, W - 1);
  int y1 = imin(y0 + 1, H - 1);
  float wx = u - (float)x0;
  float wy = v - (float)y0;
  const v4f* b = (const v4f*)base;               // 2 x v4f per (y,x) site
  int r0 = y0 << logW, r1 = y1 << logW;
  int i00 = (r0 + x0) * 2, i10 = (r0 + x1) * 2;
  int i01 = (r1 + x0) * 2, i11 = (r1 + x1) * 2;
  v4f a0 = b[i00], a1 = b[i00 + 1];
  v4f b0 = b[i10], b1 = b[i10 + 1];
  v4f c0 = b[i01], c1 = b[i01 + 1];
  v4f d0 = b[i11], d1 = b[i11 + 1];
  float w00 = (1.0f - wx) * (1.0f - wy);
  float w10 = wx * (1.0f - wy);
  float w01 = (1.0f - wx) * wy;
  float w11 = wx * wy;
  F8 r;
  r.lo = a0 * w00 + b0 * w10 + c0 * w01 + d0 * w11;
  r.hi = a1 * w00 + b1 * w10 + c1 * w01 + d1 * w11;
  return r;
}

// (8, H, W) channel-major -> (H, W, 8) channel-last into workspace.
__global__ __launch_bounds__(256) void transpose_plane(
    const float* __restrict__ src, float* __restrict__ dst,
    int logW, int logH) {
  int n = 8 << (logW + logH);
  int e = blockIdx.x * 256 + threadIdx.x;
  if (e >= n) return;
  int c  = e & 7;
  int xy = e >> 3;
  int x  = xy & ((1 << logW) - 1);
  int y  = xy >> logW;
  dst[e] = src[(c << (logW + logH)) + (y << logW) + x];
}

// Thread t of block b: point pt = b*64 + (t>>2), scale s = t&3.
// Output element g = pt*4 + s holds cols [s*8, s*8+8) of the (N,32) result.
__global__ __launch_bounds__(256) void planes_fused(
    const float* __restrict__ pts, const float* __restrict__ ws,
    float* __restrict__ out, int npts) {
  __shared__ float spts[256];   // ONLY shared alloc -> LDS byte offset 0 for the TDM D#
  (void)spts;                   // written by the TDM engine, read via asm ds_load
  const int tid = threadIdx.x;
  const int ptBase = blockIdx.x * 64;

  if (tid < 32) {  // wave 0 issues the Tensor Data Mover DMA (EXEC-independent, per-wave)
    unsigned long long ga = (unsigned long long)(pts + (size_t)ptBase * 4);
    unsigned nelem = (unsigned)imin(256, (npts - ptBase) * 4);  // tail -> OOB reads zero
    v4u g0;
    g0.x = 1u;                                            // count=1 (valid descriptor)
    g0.y = 0u;                                            // lds_addr = 0 (spts)
    g0.z = (unsigned)(ga & 0xFFFFFFFFu);                  // global_addr[31:0]
    g0.w = (unsigned)((ga >> 32) & 0x01FFFFFFu) | 0x80000000u;  // addr[56:32] | type=2
    v8i g1;
    g1[0] = (int)(2u << 16);                              // data_size=4B, no flags
    g1[1] = (int)((nelem & 0xFFFFu) << 16);               // tensor_dim0[15:0]
    g1[2] = (int)(((nelem >> 16) & 0xFFFFu) | (1u << 16));// tensor_dim0[31:16] | tensor_dim1=1
    g1[3] = (int)((nelem & 0xFFFFu) << 16);               // tensor_dim1 hi=0 | tile_dim0
    g1[4] = 1;                                            // tile_dim1=1, tile_dim2=0
    g1[5] = (int)nelem;                                   // tensor_dim0_stride lo32
    g1[6] = 0;                                            // stride0 hi | stride1 lo
    g1[7] = 0;                                            // stride1 hi
    v4i zz4 = {0, 0, 0, 0};
    v8i zz8 = {0, 0, 0, 0, 0, 0, 0, 0};
    __builtin_amdgcn_tensor_load_to_lds(g0, g1, zz4, zz4, zz8, 0);
    __builtin_amdgcn_s_wait_tensorcnt(0);
  }
  __syncthreads();

  const int lpt = tid >> 2;
  const int s   = tid & 3;
  const int pt  = ptBase + lpt;

  // Read (x,y,z,t) of this thread's point from LDS. Must be an opaque asm load:
  // the compiler cannot see the TDM engine's writes to spts and would otherwise
  // fold these reads to undef (round-2 histogram showed ds=0).
  v4f c4;
  {
    unsigned lds_addr = (unsigned)(lpt * 16);   // spts is at LDS byte offset 0
    asm volatile("ds_load_b128 %0, %1\n\t"
                 "s_wait_dscnt 0x0"
                 : "=v"(c4) : "v"(lds_addr));
  }
  if (pt >= npts) return;
  const float x  = c4.x;
  const float y  = c4.y;
  const float z  = c4.z;
  const float tt = c4.w;

  const int logR = 5 + s;                               // R = 32<<s
  const size_t base = (s == 0) ? 0u : (s == 1) ? 30720u
                    : (s == 2) ? 141312u : 559104u;     // cumulative f32 offsets per scale
  const size_t sq = (size_t)8 << (2 * logR);            // 8*R*R   (spatial plane)
  const size_t tp = (size_t)64 << logR;                 // 8*8*R   (time plane)
  const float* p01 = ws + base;        // combo (0,1): u=x, v=y
  const float* p02 = p01 + sq;         // combo (0,2): u=x, v=z
  const float* p03 = p02 + sq;         // combo (0,3): u=x, v=t  (H=8)
  const float* p12 = p03 + tp;         // combo (1,2): u=y, v=z
  const float* p13 = p12 + sq;         // combo (1,3): u=y, v=t  (H=8)
  const float* p23 = p13 + tp;         // combo (2,3): u=z, v=t  (H=8)

  F8 st = bilerp8(p01, logR, logR, x, y);
  F8 tm = bilerp8(p02, logR, logR, x, z);
  st.lo *= tm.lo; st.hi *= tm.hi;
  tm = bilerp8(p12, logR, logR, y, z);
  st.lo *= tm.lo; st.hi *= tm.hi;

  F8 dn = bilerp8(p03, logR, 3, x, tt);
  tm = bilerp8(p13, logR, 3, y, tt);
  dn.lo *= tm.lo; dn.hi *= tm.hi;
  tm = bilerp8(p23, logR, 3, z, tt);
  dn.lo *= tm.lo; dn.hi *= tm.hi;

  // Streaming 512MB result, never re-read: non-temporal stores keep the
  // L2 free for the 8.3MB gather-hot plane workspace.
  const size_t g = (size_t)pt * 4 + s;
  v4f* o = (v4f*)out;
  __builtin_nontemporal_store(st.lo, &o[g * 2]);       // static: [pt*32 + s*8 ...]
  __builtin_nontemporal_store(st.hi, &o[g * 2 + 1]);
  const size_t dOff = (size_t)npts * 8;                // dyn half at npts*32 floats
  __builtin_nontemporal_store(dn.lo, &o[dOff + g * 2]);
  __builtin_nontemporal_store(dn.hi, &o[dOff + g * 2 + 1]);
}

extern "C" void kernel_launch(void* const* d_in, const int* in_sizes, int n_in,
                              void* d_out, int out_size, void* d_ws, size_t ws_size,
                              hipStream_t stream) {
  const float* pts = (const float*)d_in[0];
  float* ws  = (float*)d_ws;   // needs ~8.4 MB: 2,181,120 floats
  float* out = (float*)d_out;
  const int npts = in_sizes[0] / 4;

  // Re-tile all 24 planes to channel-last (H,W,8) in workspace.
  size_t off = 0;
  for (int i = 0; i < 24; ++i) {
    const int s = i / 6, j = i % 6;
    const int logR = 5 + s;
    const int logW = logR;
    const int logH = (j == 0 || j == 1 || j == 3) ? logR : 3;  // time planes: H=8
    const int n = 8 << (logW + logH);
    transpose_plane<<<(n + 255) / 256, 256, 0, stream>>>(
        (const float*)d_in[1 + i], ws + off, logW, logH);
    off += (size_t)n;
  }

  const int nblocks = (npts + 63) / 64;
  planes_fused<<<nblocks, 256, 0, stream>>>(pts, ws, out, npts);
}